// Cell_to_Entity_78735340470739
// MI455X (gfx1250) — compile-verified
//
#include <hip/hip_runtime.h>

typedef __attribute__((ext_vector_type(2))) float v2f;
typedef __attribute__((ext_vector_type(4))) float v4f;
typedef __attribute__((ext_vector_type(8))) float v8f;
typedef int v4i_vs __attribute__((vector_size(4 * sizeof(int))));  // matches builtin param

#define GLOBAL_AS __attribute__((address_space(1)))
#define LDS_AS    __attribute__((address_space(3)))

constexpr int Bb = 4, Ss = 512, Ee = 64, Hh = 1024;
constexpr int R      = Bb * Ss;     // 2048 (b,s) rows
constexpr int TILE_R = 16;          // rows per WMMA tile
constexpr int WAVES  = 8;          // waves per block, one entity each
constexpr int HC     = 32;          // h-chunk per LDS stage (floats)
constexpr int PITCH  = HC + 4;      // LDS row pitch (dwords): 36 -> conflict-free A reads
constexpr int NCHUNK = Hh / HC;     // 32 chunks over H
constexpr int NBUF   = 3;           // LDS buffers per wave (prefetch depth 2)

// ---- CDNA5 async global->LDS copy (ASYNCcnt-tracked), 16B per lane ----
__device__ __forceinline__ void async_copy_b128(const float* gsrc, float* ldst) {
    __builtin_amdgcn_global_load_async_to_lds_b128(
        (GLOBAL_AS v4i_vs*)(uintptr_t)gsrc,
        (LDS_AS v4i_vs*)(unsigned)(uintptr_t)ldst,
        /*offset=*/0, /*cpol=*/0);
}

// s_wait_asynccnt as a *memory ordering point*: the builtin form is not a
// scheduling barrier, and round-3 asm showed ds_loads hoisted above it.
// The "memory" clobber pins all LDS/global accesses on the correct side.
template <int N>
__device__ __forceinline__ void wait_async() {
    asm volatile("s_wait_asynccnt %0" :: "n"(N) : "memory");
}

__global__ __launch_bounds__(256)
void cell2entity_wmma_kernel(const float* __restrict__ X,
                             const float* __restrict__ W,
                             const float* __restrict__ bias,
                             float* __restrict__ out) {
    // per-wave triple-buffered staging: 8 waves * 3 * 16 * 36 dwords = 55296 B
    __shared__ float lds[WAVES * NBUF * TILE_R * PITCH];

    const int tid  = threadIdx.x;
    const int wave = tid >> 5;
    const int lane = tid & 31;
    const int rt   = blockIdx.x;                  // 0..127 row tile
    const int e    = blockIdx.y * WAVES + wave;   // 0..63 entity
    const int r0   = rt * TILE_R;

    float* bufs = &lds[wave * NBUF * TILE_R * PITCH];

    // X[((r0+row)*E + e)*H + h]; row stride (fixed e) = E*H floats
    const size_t rowStride = (size_t)Ee * Hh;     // 65536
    const float* Xbase = X + ((size_t)r0 * Ee + (size_t)e) * Hh;
    const float* Wrow  = W + (size_t)e * Hh;

    // loader mapping: 4 rows x 128B per async b128 instruction
    const int ldRow = lane >> 3;                  // 0..3 (+ 4*i)
    const int ldCol = (lane & 7) * 4;             // 0,4,...,28
    // A/B fragment mapping (16x4 fp32 layout: lanes 0-15 K={0,1}, lanes 16-31 K={2,3})
    const int mRow  = lane & 15;
    const int kOff  = (lane >> 4) * 2;

    v8f c = {};           // 16x16 fp32 accumulator (8 VGPRs)

    // stage chunk `ch` into LDS buffer ch % NBUF via async DMA (4 ops/chunk)
    auto stage = [&](int ch) {
        float* dst = bufs + (ch % NBUF) * TILE_R * PITCH;
        const float* src = Xbase + ch * HC;
        #pragma unroll
        for (int i = 0; i < 4; ++i) {
            const int row = i * 4 + ldRow;
            async_copy_b128(src + (size_t)row * rowStride + ldCol,
                            dst + row * PITCH + ldCol);
        }
    };

    // prologue: two chunks in flight
    stage(0);
    stage(1);

    for (int ch = 0; ch < NCHUNK; ++ch) {
        if (ch + 2 < NCHUNK) stage(ch + 2);

        // async loads complete in order: retire chunk `ch`'s 4 ops
        if (ch + 2 < NCHUNK)      wait_async<8>();
        else if (ch + 1 < NCHUNK) wait_async<4>();
        else                      wait_async<0>();

        const float* cur = bufs + (ch % NBUF) * TILE_R * PITCH;
        const int h0 = ch * HC;

        // 8 x V_WMMA_F32_16X16X4_F32 over this 16x32 tile
        #pragma unroll
        for (int kc = 0; kc < HC / 4; ++kc) {
            v2f a  = *(const v2f*)(cur + mRow * PITCH + kc * 4 + kOff);
            v2f bf = *(const v2f*)(Wrow + h0 + kc * 4 + kOff);  // W replicated over N
            c = __builtin_amdgcn_wmma_f32_16x16x4_f32(
                    /*neg_a=*/false, a, /*neg_b=*/false, bf,
                    /*c_mod=*/(short)0, c, /*reuse_a=*/false, /*reuse_b=*/false);
        }
    }

    // D layout: VGPR v, lanes 0-15 -> M=v, lanes 16-31 -> M=v+8; all N equal.
    const float bv = bias[e];
    if (lane == 0 || lane == 16) {
        const int mBase = (lane >> 4) * 8;
        #pragma unroll
        for (int v = 0; v < 8; ++v) {
            out[(size_t)(r0 + mBase + v) * Ee + e] = c[v] + bv;
        }
    }
}

extern "C" void kernel_launch(void* const* d_in, const int* in_sizes, int n_in,
                              void* d_out, int out_size, void* d_ws, size_t ws_size,
                              hipStream_t stream) {
    const float* X  = (const float*)d_in[0];   // (B,S,E,H) fp32
    const float* W  = (const float*)d_in[1];   // (E,H)     fp32
    const float* bb = (const float*)d_in[2];   // (E,)      fp32
    float* out = (float*)d_out;                // (B,S,E)   fp32

    dim3 grid(R / TILE_R, Ee / WAVES);         // (128, 8)
    dim3 block(256);                           // 8 waves
    hipLaunchKernelGGL(cell2entity_wmma_kernel, grid, block, 0, stream,
                       X, W, bb, out);
}